// PAMPJPE_49237505081776
// MI455X (gfx1250) — compile-verified
//
#include <hip/hip_runtime.h>
#include <hip/hip_bf16.h>

namespace {

constexpr int kJ     = 14;
constexpr int kFPE   = 3 * kJ;          // 42 floats per element (168 B)
constexpr int kElems = 128;             // elements per workgroup
constexpr int kBlock = 128;             // threads per workgroup = 4 wave32
constexpr int kTileF = kElems * kFPE;   // 5376 floats per tile (21504 B)
constexpr int kTileV4 = kTileF / 4;     // 1344 float4 per tile

__device__ __forceinline__ float det3(float a, float b, float c,
                                      float d, float e, float f,
                                      float g, float h, float i) {
  return a * (e * i - f * h) - b * (d * i - f * g) + c * (d * h - e * g);
}

#if defined(__gfx1250__) && __has_builtin(__builtin_amdgcn_global_load_async_to_lds_b128)
#define PAMPJPE_ASYNC_BUILTIN 1
#endif

#if defined(__gfx1250__)
typedef int v4i_pam __attribute__((ext_vector_type(4)));
typedef __attribute__((address_space(1))) v4i_pam* pam_gptr;
typedef __attribute__((address_space(3))) v4i_pam* pam_lptr;
#endif

// Stage 16 bytes from global memory into LDS via the CDNA5 async-to-LDS path.
__device__ __forceinline__ void stage_b128(const float4* src, float4* dstLds) {
#if defined(PAMPJPE_ASYNC_BUILTIN)
  __builtin_amdgcn_global_load_async_to_lds_b128(
      (pam_gptr)src, (pam_lptr)dstLds, /*imm offset*/ 0, /*cpol*/ 0);
#elif defined(__gfx1250__)
  unsigned ldsOff = (unsigned)(pam_lptr)dstLds;
  asm volatile("global_load_async_to_lds_b128 %0, %1, off"
               :: "v"(ldsOff), "v"(src)
               : "memory");
#else
  *dstLds = *src;
#endif
}

__device__ __forceinline__ void stage_wait() {
#if defined(__gfx1250__)
#if __has_builtin(__builtin_amdgcn_s_wait_asynccnt)
  __builtin_amdgcn_s_wait_asynccnt(0);
#else
  asm volatile("s_wait_asynccnt 0x0" ::: "memory");
#endif
#endif
  __syncthreads();
}

__global__ __launch_bounds__(kBlock)
void pa_mpjpe_kernel(const float* __restrict__ pred,
                     const float* __restrict__ gt,
                     float* __restrict__ out, int B) {
  __shared__ float sP[kTileF];
  __shared__ float sG[kTileF];

  const int base   = blockIdx.x * kElems;
  const int remain = B - base;
  const int nE     = remain < kElems ? remain : kElems;

  if (nE == kElems) {
    // Full tile: contiguous, 16B-aligned chunk -> coalesced async B128 staging.
    const float4* gp = (const float4*)(pred + (size_t)base * kFPE);
    const float4* gg = (const float4*)(gt   + (size_t)base * kFPE);
    float4* lp = (float4*)sP;
    float4* lg = (float4*)sG;
    for (int i = threadIdx.x; i < kTileV4; i += kBlock) {
      stage_b128(gp + i, lp + i);
      stage_b128(gg + i, lg + i);
    }
    stage_wait();
  } else {
    // Ragged tail: plain scalar staging.
    const int nF = nE * kFPE;
    for (int i = threadIdx.x; i < nF; i += kBlock) {
      sP[i] = pred[(size_t)base * kFPE + i];
      sG[i] = gt[(size_t)base * kFPE + i];
    }
    __syncthreads();
  }

  const int t    = threadIdx.x;
  const int elem = base + t;
  if (t >= nE) return;

  // Per-thread element records in LDS; stride 42 dwords across lanes ->
  // 32 lanes map to 32 distinct banks (gcd(42,64)=2, 21 coprime 32): conflict-free.
  const float* p = sP + t * kFPE;
  const float* g = sG + t * kFPE;

  // ---- Pass 1: raw sums -> means, var1/var2, cross-covariance K ----
  float sx0 = 0.f, sx1 = 0.f, sx2 = 0.f, sy0 = 0.f, sy1 = 0.f, sy2 = 0.f;
  float sxx = 0.f, syy = 0.f;
  float k00 = 0.f, k01 = 0.f, k02 = 0.f;
  float k10 = 0.f, k11 = 0.f, k12 = 0.f;
  float k20 = 0.f, k21 = 0.f, k22 = 0.f;
#pragma unroll
  for (int j = 0; j < kJ; ++j) {
    const float x0 = p[3 * j + 0], x1 = p[3 * j + 1], x2 = p[3 * j + 2];
    const float y0 = g[3 * j + 0], y1 = g[3 * j + 1], y2 = g[3 * j + 2];
    sx0 += x0; sx1 += x1; sx2 += x2;
    sy0 += y0; sy1 += y1; sy2 += y2;
    sxx += x0 * x0 + x1 * x1 + x2 * x2;
    syy += y0 * y0 + y1 * y1 + y2 * y2;
    k00 += x0 * y0; k01 += x0 * y1; k02 += x0 * y2;
    k10 += x1 * y0; k11 += x1 * y1; k12 += x1 * y2;
    k20 += x2 * y0; k21 += x2 * y1; k22 += x2 * y2;
  }

  const float invJ = 1.0f / (float)kJ;
  const float mu1x = sx0 * invJ, mu1y = sx1 * invJ, mu1z = sx2 * invJ;
  const float mu2x = sy0 * invJ, mu2y = sy1 * invJ, mu2z = sy2 * invJ;
  k00 -= sx0 * sy0 * invJ; k01 -= sx0 * sy1 * invJ; k02 -= sx0 * sy2 * invJ;
  k10 -= sx1 * sy0 * invJ; k11 -= sx1 * sy1 * invJ; k12 -= sx1 * sy2 * invJ;
  k20 -= sx2 * sy0 * invJ; k21 -= sx2 * sy1 * invJ; k22 -= sx2 * sy2 * invJ;
  const float var1 = sxx - (sx0 * sx0 + sx1 * sx1 + sx2 * sx2) * invJ;
  const float var2 = syy - (sy0 * sy0 + sy1 * sy1 + sy2 * sy2) * invJ;

  // ---- Horn 4x4 N from K (symmetric) ----
  const float n00 = k00 + k11 + k22;
  const float n01 = k12 - k21;
  const float n02 = k20 - k02;
  const float n03 = k01 - k10;
  const float n11 = k00 - k11 - k22;
  const float n12 = k01 + k10;
  const float n13 = k20 + k02;
  const float n22 = k11 - k00 - k22;
  const float n23 = k12 + k21;
  const float n33 = k22 - k00 - k11;

  // ---- QCP characteristic polynomial: x^4 + C2 x^2 + C1 x + C0 ----
  const float fro2 = k00 * k00 + k01 * k01 + k02 * k02 +
                     k10 * k10 + k11 * k11 + k12 * k12 +
                     k20 * k20 + k21 * k21 + k22 * k22;
  const float C2 = -2.0f * fro2;
  const float C1 = -8.0f * det3(k00, k01, k02, k10, k11, k12, k20, k21, k22);
  const float m0 = det3(n11, n12, n13, n12, n22, n23, n13, n23, n33);
  const float m1 = det3(n01, n12, n13, n02, n22, n23, n03, n23, n33);
  const float m2 = det3(n01, n11, n13, n02, n12, n23, n03, n13, n33);
  const float m3 = det3(n01, n11, n12, n02, n12, n22, n03, n13, n23);
  const float C0 = n00 * m0 - n01 * m1 + n02 * m2 - n03 * m3;

  // ---- Newton for lambda_max (monotone from lam0 >= lam_max) ----
  float lam = 0.5f * (var1 + var2);
#pragma unroll
  for (int it = 0; it < 12; ++it) {
    const float l2 = lam * lam;
    const float P  = (l2 + C2) * l2 + C1 * lam + C0;
    const float dP = (4.0f * l2 + 2.0f * C2) * lam + C1;
    const float den = fabsf(dP) > 1e-18f ? dP : copysignf(1e-18f, dP);
    lam -= P / den;
  }

  // ---- Eigenvector of N at lam via adjugate cofactors (with fallback) ----
  const float a11 = n00 - lam, a22 = n11 - lam, a33 = n22 - lam, a44 = n33 - lam;
  const float a12 = n01, a13 = n02, a14 = n03, a23 = n12, a24 = n13, a34 = n23;
  float q0 =  det3(a22, a23, a24, a23, a33, a34, a24, a34, a44);
  float q1 = -det3(a12, a23, a24, a13, a33, a34, a14, a34, a44);
  float q2 =  det3(a12, a22, a24, a13, a23, a34, a14, a24, a44);
  float q3 = -det3(a12, a22, a23, a13, a23, a33, a14, a24, a34);
  float qq = q0 * q0 + q1 * q1 + q2 * q2 + q3 * q3;
  if (qq < 1e-11f) {
    q0 = -det3(a12, a13, a14, a23, a33, a34, a24, a34, a44);
    q1 =  det3(a11, a13, a14, a13, a33, a34, a14, a34, a44);
    q2 = -det3(a11, a12, a14, a13, a23, a34, a14, a24, a44);
    q3 =  det3(a11, a12, a13, a13, a23, a33, a14, a24, a34);
    qq = q0 * q0 + q1 * q1 + q2 * q2 + q3 * q3;
    if (qq < 1e-11f) { q0 = 1.0f; q1 = q2 = q3 = 0.0f; qq = 1.0f; }
  }
  const float qn = rsqrtf(qq);
  const float w = q0 * qn, x = q1 * qn, y = q2 * qn, z = q3 * qn;

  const float r00 = 1.0f - 2.0f * (y * y + z * z);
  const float r01 = 2.0f * (x * y - w * z);
  const float r02 = 2.0f * (x * z + w * y);
  const float r10 = 2.0f * (x * y + w * z);
  const float r11 = 1.0f - 2.0f * (x * x + z * z);
  const float r12 = 2.0f * (y * z - w * x);
  const float r20 = 2.0f * (x * z - w * y);
  const float r21 = 2.0f * (y * z + w * x);
  const float r22 = 1.0f - 2.0f * (x * x + y * y);

  const float scl = lam / fmaxf(var1, 1e-12f);

  // ---- Pass 2: per-joint error ||s*R*(p-mu1) - (g-mu2)||, mean over joints ----
  float acc = 0.0f;
#pragma unroll
  for (int j = 0; j < kJ; ++j) {
    const float x0 = p[3 * j + 0] - mu1x;
    const float x1 = p[3 * j + 1] - mu1y;
    const float x2 = p[3 * j + 2] - mu1z;
    const float rx = r00 * x0 + r01 * x1 + r02 * x2;
    const float ry = r10 * x0 + r11 * x1 + r12 * x2;
    const float rz = r20 * x0 + r21 * x1 + r22 * x2;
    const float e0 = scl * rx - (g[3 * j + 0] - mu2x);
    const float e1 = scl * ry - (g[3 * j + 1] - mu2y);
    const float e2 = scl * rz - (g[3 * j + 2] - mu2z);
    acc += sqrtf(e0 * e0 + e1 * e1 + e2 * e2);
  }
  out[elem] = acc * invJ;
}

}  // namespace

extern "C" void kernel_launch(void* const* d_in, const int* in_sizes, int n_in,
                              void* d_out, int out_size, void* d_ws, size_t ws_size,
                              hipStream_t stream) {
  (void)n_in; (void)d_ws; (void)ws_size; (void)out_size;
  const float* pred = (const float*)d_in[0];
  const float* gt   = (const float*)d_in[1];
  float* out        = (float*)d_out;
  const int B = in_sizes[0] / kFPE;   // 262144
  const int grid = (B + kElems - 1) / kElems;
  hipLaunchKernelGGL(pa_mpjpe_kernel, dim3(grid), dim3(kBlock), 0, stream,
                     pred, gt, out, B);
}